// GraphConv_21689584844834
// MI455X (gfx1250) — compile-verified
//
#include <hip/hip_runtime.h>

#define N_NODES 100000
#define N_EDGES 1600000
#define DIM 128
#define ROW_TILES (N_NODES / 16)   // 6250, exact
#define LW_STRIDE 144              // float2 stride per k-pair row: 288 dwords == 32 mod 64
                                   // -> h=0 and h=1 half-waves hit disjoint LDS bank halves

typedef __attribute__((ext_vector_type(2))) float v2f;
typedef __attribute__((ext_vector_type(8))) float v8f;

// ---------------------------------------------------------------------------
// Kernel 1: zero the aggregation buffer (N*D floats) with wide stores.
// ---------------------------------------------------------------------------
__global__ void gc_zero_ws(float4* __restrict__ ws) {
    int i = blockIdx.x * blockDim.x + threadIdx.x;   // grid sized exactly
    ws[i] = make_float4(0.f, 0.f, 0.f, 0.f);
}

// ---------------------------------------------------------------------------
// Kernel 2: edge gather + scale + scatter-add.
// One wave (32 lanes) per edge; lane l covers columns [4l, 4l+4).
// agg (51.2 MB) fits in the 192 MB L2, so the f32 atomics resolve in L2.
// ---------------------------------------------------------------------------
__global__ void __launch_bounds__(256)
gc_edge_scatter(const float* __restrict__ x,
                const long long* __restrict__ edge_index,   // [2, E] int64
                const float* __restrict__ edge_weight,      // [E]
                float* __restrict__ agg) {
    int tid  = blockIdx.x * blockDim.x + threadIdx.x;
    int e    = tid >> 5;
    int lane = tid & 31;
    if (e >= N_EDGES) return;

    int src = (int)edge_index[e];               // row 0 of edge_index
    int dst = (int)edge_index[N_EDGES + e];     // row 1 of edge_index
    float w = edge_weight[e];

    // coalesced: wave reads one contiguous 512B row of x
    const float4 v = ((const float4*)(x + (size_t)src * DIM))[lane];

    float* base = agg + (size_t)dst * DIM + lane * 4;
    atomicAdd(base + 0, v.x * w);
    atomicAdd(base + 1, v.y * w);
    atomicAdd(base + 2, v.z * w);
    atomicAdd(base + 3, v.w * w);
}

// ---------------------------------------------------------------------------
// Kernel 3: out = agg @ W2 + b2 using V_WMMA_F32_16X16X4_F32.
// 256 threads = 8 waves per block; each wave owns one 16-row tile and all
// 8 column tiles (8 x v8f accumulators).
//
// W2 is staged into LDS in k-pair-interleaved form:
//   lW[q * LW_STRIDE + n] = ( W2[2q][n], W2[2q+1][n] )        q = 0..63
// so every B fragment is ONE contiguous 8-byte LDS load that lands directly
// in an even-aligned VGPR pair for WMMA src1 (no cross-register repacking),
// and the padded stride makes the two half-waves use disjoint bank halves.
//
// VGPR layouts (ISA 7.12.2), lane = 16h + m, h in {0,1}, m in 0..15:
//   A 16x4 f32 : a = ( A[m][k0+2h], A[m][k0+2h+1] )           (global_load_b64)
//   B  4x16 f32: b = ( B[k0+2h][n], B[k0+2h+1][n] ), n = m    (ds_load_b64)
//   C/D 16x16  : acc[i] = D[i + 8h][n]
// ---------------------------------------------------------------------------
__global__ void __launch_bounds__(256)
gc_gemm_wmma(const float* __restrict__ agg,
             const float* __restrict__ W2,
             const float* __restrict__ b2,
             float* __restrict__ out) {
    __shared__ float2 lW[64 * LW_STRIDE];    // 72 KB of the 320 KB/WGP LDS

    // cooperative, coalesced fill: idx enumerates (row-pair q, float4 col group c4)
    for (int idx = threadIdx.x; idx < 64 * 32; idx += 256) {
        int q  = idx >> 5;                   // k-pair: rows 2q, 2q+1 of W2
        int c4 = idx & 31;                   // columns 4*c4 .. 4*c4+3
        float4 r0 = ((const float4*)(W2 + (size_t)(2 * q)     * DIM))[c4];
        float4 r1 = ((const float4*)(W2 + (size_t)(2 * q + 1) * DIM))[c4];
        float2* dst = lW + q * LW_STRIDE + c4 * 4;
        dst[0] = make_float2(r0.x, r1.x);
        dst[1] = make_float2(r0.y, r1.y);
        dst[2] = make_float2(r0.z, r1.z);
        dst[3] = make_float2(r0.w, r1.w);
    }
    __syncthreads();

    int wave = threadIdx.x >> 5;
    int lane = threadIdx.x & 31;
    int tile = blockIdx.x * 8 + wave;
    if (tile >= ROW_TILES) return;           // whole-wave exit: EXEC stays all-1s

    int r0 = tile * 16;
    int m  = lane & 15;
    int h  = lane >> 4;

    v8f acc[8] = {};                         // 64 VGPRs of accumulators

    const float* arow = agg + (size_t)(r0 + m) * DIM;   // A row for this lane

    for (int k0 = 0; k0 < DIM; k0 += 4) {
        v2f a;
        a.x = arow[k0 + 2 * h];              // contiguous pair -> global_load_b64
        a.y = arow[k0 + 2 * h + 1];

        const float2* brow = lW + ((k0 >> 1) + h) * LW_STRIDE + m;
#pragma unroll
        for (int t = 0; t < 8; ++t) {        // 8 column tiles of 16
            float2 bp = brow[t * 16];        // single ds_load_b64, even-pair ready
            v2f b;
            b.x = bp.x;
            b.y = bp.y;
            acc[t] = __builtin_amdgcn_wmma_f32_16x16x4_f32(
                /*neg_a=*/false, a, /*neg_b=*/false, b,
                /*c_mod=*/(short)0, acc[t],
                /*reuse_a=*/false, /*reuse_b=*/false);
        }
    }

    // store D + bias (half-wave coalesced b32 stores)
#pragma unroll
    for (int t = 0; t < 8; ++t) {
        float bias = b2[t * 16 + m];
#pragma unroll
        for (int i = 0; i < 8; ++i) {
            out[(size_t)(r0 + i + 8 * h) * DIM + t * 16 + m] = acc[t][i] + bias;
        }
    }
}

// ---------------------------------------------------------------------------
extern "C" void kernel_launch(void* const* d_in, const int* in_sizes, int n_in,
                              void* d_out, int out_size, void* d_ws, size_t ws_size,
                              hipStream_t stream) {
    const float*     x  = (const float*)d_in[0];
    const long long* ei = (const long long*)d_in[1];   // int64 [2,E]
    const float*     ew = (const float*)d_in[2];
    // d_in[3] = W1, d_in[4] = b1 : dead code in the reference, skipped.
    const float*     W2 = (const float*)d_in[5];
    const float*     b2 = (const float*)d_in[6];
    float* out = (float*)d_out;
    float* agg = (float*)d_ws;                         // N*D*4 = 51.2 MB scratch

    // 1) agg = 0
    gc_zero_ws<<<(N_NODES * DIM) / (4 * 256), 256, 0, stream>>>((float4*)agg);

    // 2) scatter: one wave per edge -> E*32 threads (exactly 200000 blocks)
    gc_edge_scatter<<<(N_EDGES * 32) / 256, 256, 0, stream>>>(x, ei, ew, agg);

    // 3) GEMM with fp32 WMMA
    gc_gemm_wmma<<<(ROW_TILES + 7) / 8, 256, 0, stream>>>(agg, W2, b2, out);
}